// KalmanVAE_74517682586471
// MI455X (gfx1250) — compile-verified
//
#include <hip/hip_runtime.h>

#define T_STEPS 512
#define BATCH   512
#define LAT     8
#define OBSD    16

typedef float v2f __attribute__((ext_vector_type(2)));
typedef float v8f __attribute__((ext_vector_type(8)));

// D = A(16x4) * B(4x16) + C, f32 WMMA. NEG/NEG_HI unused for f32 (C-mods only).
__device__ __forceinline__ v8f wmma4(v2f a, v2f b, v8f c) {
  return __builtin_amdgcn_wmma_f32_16x16x4_f32(false, a, false, b, (short)0, c,
                                               false, false);
}

// ---- layout helpers -------------------------------------------------------
// Canonical register format: WMMA C/D layout for a 16x16 f32 matrix:
//   lane<16 : vgpr v holds (row=v,     col=lane)
//   lane>=16: vgpr v holds (row=v+8,   col=lane-16)
// LDS staging tile is row-major with stride 17 (bank-conflict padding).

__device__ __forceinline__ void cd_to_lds(float* lds, v8f d, int lane,
                                          float scale) {
  const int h = (lane >> 4) & 1;
  const int col = lane & 15;
#pragma unroll
  for (int v = 0; v < 8; ++v) lds[(v + 8 * h) * 17 + col] = d[v] * scale;
}

// A-operand slice s covers K = 4s..4s+3:
//   lane<16 : a = {M[l][4s],   M[l][4s+1]}   (M rows = lane)
//   lane>=16: a = {M[l-16][4s+2], M[l-16][4s+3]}
__device__ __forceinline__ v2f lds_a_slice(const float* lds, int lane, int s) {
  const int M = lane & 15;
  const int k = 4 * s + ((lane >> 3) & 2);
  v2f a;
  a.x = lds[M * 17 + k];
  a.y = lds[M * 17 + k + 1];
  return a;
}

// B-operand slice s: b = {M[k][n], M[k+1][n]} with n=lane&15, k=4s+2*(lane>=16)
__device__ __forceinline__ v2f lds_b_slice(const float* lds, int lane, int s) {
  const int n = lane & 15;
  const int k = 4 * s + ((lane >> 3) & 2);
  v2f b;
  b.x = lds[k * 17 + n];
  b.y = lds[(k + 1) * 17 + n];
  return b;
}

// A-slice of X straight from per-lane row registers (lane owns row lane&15).
// NOTE: identical values also serve as the B-slice of X^T.
template <int S>
__device__ __forceinline__ v2f reg_slice(const float (&row)[8], int lane) {
  const bool hi = (lane & 16) != 0;
  v2f a;
  a.x = hi ? row[4 * S + 2] : row[4 * S + 0];
  a.y = hi ? row[4 * S + 3] : row[4 * S + 1];
  return a;
}

// Gauss-Jordan inverse of the NxN top-left block (N<=16), no pivoting
// (S = R + C Sig C^T is SPD with unit-dominated diagonal).
// Column-per-lane layout of augmented [S | I]:
//   lanes 0..15  own columns of S      (loaded from ldsM)
//   lanes 16..31 own columns of I -> become columns of S^-1, stored to ldsInv
// Cross-lane traffic per pivot k: N wave-uniform readlane broadcasts.
template <int N>
__device__ __forceinline__ void gj_invert(const float* ldsM, float* ldsInv,
                                          int lane) {
  float col[N];
  const int c = lane & 15;
  const bool inv = lane >= 16;
#pragma unroll
  for (int i = 0; i < N; ++i)
    col[i] = inv ? ((i == c) ? 1.f : 0.f) : ldsM[i * 17 + c];
#pragma unroll
  for (int k = 0; k < N; ++k) {
    const float piv = __shfl(col[k], k, 32);           // S[k][k] -> readlane
    const float ip = __builtin_amdgcn_rcpf(piv);
    const float pr = col[k] * ip;                      // scaled pivot-row elem
    col[k] = pr;
#pragma unroll
    for (int i = 0; i < N; ++i) {
      if (i == k) continue;
      const float f = __shfl(col[i], k, 32);           // factor column
      col[i] = fmaf(-f, pr, col[i]);
    }
  }
  if (inv) {
#pragma unroll
    for (int i = 0; i < N; ++i) ldsInv[i * 17 + c] = col[i];
  }
}

// ---- kernel: one wave32 per batch element, forward filter + RTS smoother ---
__global__ __launch_bounds__(32) void kalman_rts_kernel(
    const float* __restrict__ obs,   // (T,B,16)
    const float* __restrict__ Ag,    // (B,T,8,8)
    const float* __restrict__ Cg,    // (B,T,16,8)
    const float* __restrict__ mu1,   // (8)
    const float* __restrict__ Sig1,  // (8,8)
    const float* __restrict__ Qg,    // (8,8)
    const float* __restrict__ Rg,    // (16,16)
    float* __restrict__ out,         // (T,B,8,9)
    float* __restrict__ ws) {        // (B,T,72) predicted mu|Sig
  __shared__ float buf0[16 * 17];
  __shared__ float buf1[16 * 17];
  __shared__ float buf2[16 * 17];

  const int lane = threadIdx.x;
  const int b = blockIdx.x;
  const int h = (lane >> 4) & 1;
  const int col = lane & 15;
  const int m = col;
  const v8f vzero = {0.f, 0.f, 0.f, 0.f, 0.f, 0.f, 0.f, 0.f};

  // constants in C/D layout (R full 16x16; Q zero-padded 8x8)
  v8f Rcd, Qcd;
#pragma unroll
  for (int v = 0; v < 8; ++v) {
    const int row = v + 8 * h;
    Rcd[v] = Rg[row * 16 + col];
    Qcd[v] = (row < 8 && col < 8) ? Qg[row * 8 + col] : 0.f;
  }

  // predicted state (mu replicated per-lane, Sig in C/D layout, zero-padded)
  float mu[8];
#pragma unroll
  for (int v = 0; v < 8; ++v) mu[v] = mu1[v];
  v8f Sig;
#pragma unroll
  for (int v = 0; v < 8; ++v) {
    const int row = v + 8 * h;
    Sig[v] = (row < 8 && col < 8) ? Sig1[row * 8 + col] : 0.f;
  }

  float muz[8];
  v8f Sz = Sig;

  // ============================ forward filter =============================
  for (int t = 0; t < T_STEPS; ++t) {
    const long bt = (long)b * T_STEPS + t;
    const float* Abt = Ag + bt * 64;
    const float* Cbt = Cg + bt * 128;
    if (t + 1 < T_STEPS) {  // pull next step's tiles toward the WGP
      __builtin_prefetch(Abt + 64 + lane * 2, 0, 1);
      __builtin_prefetch(Cbt + 128 + lane * 4, 0, 1);
    }

    // per-lane rows: Crow = C[row=col][:], Arow = A[row=m][:] (zero if m>=8)
    float Crow[8], Arow[8];
    {
      const float4 c0 = *reinterpret_cast<const float4*>(Cbt + col * 8);
      const float4 c1 = *reinterpret_cast<const float4*>(Cbt + col * 8 + 4);
      Crow[0] = c0.x; Crow[1] = c0.y; Crow[2] = c0.z; Crow[3] = c0.w;
      Crow[4] = c1.x; Crow[5] = c1.y; Crow[6] = c1.z; Crow[7] = c1.w;
      if (m < 8) {
        const float4 a0 = *reinterpret_cast<const float4*>(Abt + m * 8);
        const float4 a1 = *reinterpret_cast<const float4*>(Abt + m * 8 + 4);
        Arow[0] = a0.x; Arow[1] = a0.y; Arow[2] = a0.z; Arow[3] = a0.w;
        Arow[4] = a1.x; Arow[5] = a1.y; Arow[6] = a1.z; Arow[7] = a1.w;
      } else {
#pragma unroll
        for (int v = 0; v < 8; ++v) Arow[v] = 0.f;
      }
    }

    // innovation r_o = y_o - C[o][:] . mu   (o = col, both halves duplicate)
    float r = obs[((long)t * BATCH + b) * OBSD + col];
#pragma unroll
    for (int v = 0; v < 8; ++v) r = fmaf(-Crow[v], mu[v], r);

    // stash predicted (mu, Sig) for the smoother
    float* wsp = ws + bt * 72;
    if (lane == 0) {
#pragma unroll
      for (int v = 0; v < 8; ++v) wsp[v] = mu[v];
    }
    if (lane < 8) {
#pragma unroll
      for (int v = 0; v < 8; ++v) wsp[8 + v * 8 + col] = Sig[v];
    }

    // W = Sig * C^T            (16x16x8 contraction; B-of-C^T == A-of-C regs)
    __syncthreads();
    cd_to_lds(buf0, Sig, lane, 1.f);
    __syncthreads();
    v8f W = vzero;
    W = wmma4(lds_a_slice(buf0, lane, 0), reg_slice<0>(Crow, lane), W);
    W = wmma4(lds_a_slice(buf0, lane, 1), reg_slice<1>(Crow, lane), W);

    // S = C * W + R
    cd_to_lds(buf1, W, lane, 1.f);
    __syncthreads();
    v8f S = Rcd;
    S = wmma4(reg_slice<0>(Crow, lane), lds_b_slice(buf1, lane, 0), S);
    S = wmma4(reg_slice<1>(Crow, lane), lds_b_slice(buf1, lane, 1), S);

    // S^-1 via column-parallel Gauss-Jordan
    cd_to_lds(buf0, S, lane, 1.f);
    __syncthreads();
    gj_invert<16>(buf0, buf2, lane);
    __syncthreads();

    // K = W * S^-1             (K dim = 16)
    v8f K = vzero;
#pragma unroll
    for (int s = 0; s < 4; ++s)
      K = wmma4(lds_a_slice(buf1, lane, s), lds_b_slice(buf2, lane, s), K);

    // mu_z = mu + K r : per-lane partial K[m][o]*r_o, butterfly over o
    float part[8];
#pragma unroll
    for (int v = 0; v < 8; ++v) part[v] = K[v] * r;
#pragma unroll
    for (int off = 1; off < 16; off <<= 1) {
#pragma unroll
      for (int v = 0; v < 8; ++v) part[v] += __shfl_xor(part[v], off, 32);
    }
#pragma unroll
    for (int v = 0; v < 8; ++v) muz[v] = mu[v] + __shfl(part[v], col, 32);

    // Sig_z = Sig - K * W^T    (stage -K; B-of-W^T == A-slices of W in buf1)
    __syncthreads();
    cd_to_lds(buf0, K, lane, -1.f);
    __syncthreads();
    Sz = Sig;
#pragma unroll
    for (int s = 0; s < 4; ++s)
      Sz = wmma4(lds_a_slice(buf0, lane, s), lds_a_slice(buf1, lane, s), Sz);

    // stage filtered state into d_out[t] (smoother overwrites t < T-1)
    float* op = out + ((long)t * BATCH + b) * 8 * 9;
    if (lane == 0) {
#pragma unroll
      for (int v = 0; v < 8; ++v) op[v * 9] = muz[v];
    }
    if (lane < 8) {
#pragma unroll
      for (int v = 0; v < 8; ++v) op[v * 9 + 1 + col] = Sz[v];
    }

    // predict: Y = A*Sig_z ; Sig' = Y*A^T + Q ; mu' = A*mu_z
    __syncthreads();
    cd_to_lds(buf0, Sz, lane, 1.f);
    __syncthreads();
    v8f Y = vzero;
    Y = wmma4(reg_slice<0>(Arow, lane), lds_b_slice(buf0, lane, 0), Y);
    Y = wmma4(reg_slice<1>(Arow, lane), lds_b_slice(buf0, lane, 1), Y);
    __syncthreads();
    cd_to_lds(buf0, Y, lane, 1.f);
    __syncthreads();
    v8f Sn = Qcd;
    Sn = wmma4(lds_a_slice(buf0, lane, 0), reg_slice<0>(Arow, lane), Sn);
    Sn = wmma4(lds_a_slice(buf0, lane, 1), reg_slice<1>(Arow, lane), Sn);
    Sig = Sn;

    float val = 0.f;
#pragma unroll
    for (int v = 0; v < 8; ++v) val = fmaf(Arow[v], muz[v], val);
#pragma unroll
    for (int v = 0; v < 8; ++v) mu[v] = __shfl(val, v, 32);
  }

  // ============================ RTS smoother ===============================
  float musn[8];
#pragma unroll
  for (int v = 0; v < 8; ++v) musn[v] = muz[v];
  v8f Ssn = Sz;  // filtered state at T-1 is the final smoothed state

  for (int t = T_STEPS - 2; t >= 0; --t) {
    const long bt1 = (long)b * T_STEPS + (t + 1);
    const float* Abt = Ag + bt1 * 64;
    float Arow[8];
    if (m < 8) {
      const float4 a0 = *reinterpret_cast<const float4*>(Abt + m * 8);
      const float4 a1 = *reinterpret_cast<const float4*>(Abt + m * 8 + 4);
      Arow[0] = a0.x; Arow[1] = a0.y; Arow[2] = a0.z; Arow[3] = a0.w;
      Arow[4] = a1.x; Arow[5] = a1.y; Arow[6] = a1.z; Arow[7] = a1.w;
    } else {
#pragma unroll
      for (int v = 0; v < 8; ++v) Arow[v] = 0.f;
    }

    // predicted state at t+1
    const float* wsp = ws + bt1 * 72;
    float mup[8];
#pragma unroll
    for (int v = 0; v < 8; ++v) mup[v] = wsp[v];
    v8f P;
#pragma unroll
    for (int v = 0; v < 8; ++v) {
      const int row = v + 8 * h;
      P[v] = (row < 8 && col < 8) ? wsp[8 + row * 8 + col] : 0.f;
    }

    // filtered state at t (staged in d_out)
    float* op = out + ((long)t * BATCH + b) * 8 * 9;
    float muf[8];
#pragma unroll
    for (int v = 0; v < 8; ++v) muf[v] = op[v * 9];
    v8f F;
#pragma unroll
    for (int v = 0; v < 8; ++v) {
      const int row = v + 8 * h;
      F[v] = (row < 8 && col < 8) ? op[row * 9 + 1 + col] : 0.f;
    }

    // P^-1 (8x8)
    __syncthreads();
    cd_to_lds(buf0, P, lane, 1.f);
    __syncthreads();
    gj_invert<8>(buf0, buf2, lane);
    __syncthreads();

    // M1 = Sig_f * A^T ; J = M1 * P^-1
    cd_to_lds(buf0, F, lane, 1.f);
    __syncthreads();
    v8f M1 = vzero;
    M1 = wmma4(lds_a_slice(buf0, lane, 0), reg_slice<0>(Arow, lane), M1);
    M1 = wmma4(lds_a_slice(buf0, lane, 1), reg_slice<1>(Arow, lane), M1);
    __syncthreads();
    cd_to_lds(buf1, M1, lane, 1.f);
    __syncthreads();
    v8f J = vzero;
    J = wmma4(lds_a_slice(buf1, lane, 0), lds_b_slice(buf2, lane, 0), J);
    J = wmma4(lds_a_slice(buf1, lane, 1), lds_b_slice(buf2, lane, 1), J);

    // stage J (serves A-of-J, B-of-J^T, and J row reads) and dSig
    __syncthreads();
    cd_to_lds(buf0, J, lane, 1.f);
    v8f dS;
#pragma unroll
    for (int v = 0; v < 8; ++v) dS[v] = Ssn[v] - P[v];
    cd_to_lds(buf1, dS, lane, 1.f);
    __syncthreads();

    // Z = J * dSig
    v8f Z = vzero;
    Z = wmma4(lds_a_slice(buf0, lane, 0), lds_b_slice(buf1, lane, 0), Z);
    Z = wmma4(lds_a_slice(buf0, lane, 1), lds_b_slice(buf1, lane, 1), Z);

    // mu_s = mu_f + J * (mu_s_next - mu_p) : row-dot from staged J
    float dmu[8];
#pragma unroll
    for (int v = 0; v < 8; ++v) dmu[v] = musn[v] - mup[v];
    float sacc = 0.f;
#pragma unroll
    for (int v = 0; v < 8; ++v) sacc = fmaf(buf0[m * 17 + v], dmu[v], sacc);
#pragma unroll
    for (int v = 0; v < 8; ++v) musn[v] = muf[v] + __shfl(sacc, v, 32);

    // Sig_s = Z * J^T + Sig_f   (B-of-J^T == A-slices of J in buf0)
    __syncthreads();
    cd_to_lds(buf1, Z, lane, 1.f);
    __syncthreads();
    v8f Ss = F;
    Ss = wmma4(lds_a_slice(buf1, lane, 0), lds_a_slice(buf0, lane, 0), Ss);
    Ss = wmma4(lds_a_slice(buf1, lane, 1), lds_a_slice(buf0, lane, 1), Ss);
    Ssn = Ss;

    // overwrite d_out[t] with smoothed state
    if (lane == 0) {
#pragma unroll
      for (int v = 0; v < 8; ++v) op[v * 9] = musn[v];
    }
    if (lane < 8) {
#pragma unroll
      for (int v = 0; v < 8; ++v) op[v * 9 + 1 + col] = Ss[v];
    }
  }
}

extern "C" void kernel_launch(void* const* d_in, const int* in_sizes, int n_in,
                              void* d_out, int out_size, void* d_ws,
                              size_t ws_size, hipStream_t stream) {
  const float* obs = (const float*)d_in[0];   // (512,512,16)
  const float* A = (const float*)d_in[1];     // (512,512,8,8)
  const float* C = (const float*)d_in[2];     // (512,512,16,8)
  const float* mu1 = (const float*)d_in[3];   // (8)
  const float* Sig1 = (const float*)d_in[4];  // (8,8)
  const float* Q = (const float*)d_in[5];     // (8,8)
  const float* R = (const float*)d_in[6];     // (16,16)
  float* out = (float*)d_out;                 // (512,512,8,9)
  float* ws = (float*)d_ws;                   // (B,T,72) predicted state

  kalman_rts_kernel<<<dim3(BATCH), dim3(32), 0, stream>>>(
      obs, A, C, mu1, Sig1, Q, R, out, ws);
}